// FullScan_7507602833574
// MI455X (gfx1250) — compile-verified
//
#include <hip/hip_runtime.h>

typedef __attribute__((ext_vector_type(2))) float v2f;
typedef __attribute__((ext_vector_type(8))) float v8f;

namespace {
constexpr int Dch   = 16;
constexpr int Llen  = 1000000;
constexpr int Wlen  = 64;
constexpr int Kout  = 8;
constexpr int Pad   = 31;                    // (W-1)/2
constexpr int MidLen = Llen - Wlen + 1;      // 999937
constexpr int WavesPerBlock = 4;
constexpr int TilesPerWave  = 4;
constexpr int TilePos = 16;
constexpr int WavePos  = TilesPerWave * TilePos;        // 64
constexpr int BlockPos = WavesPerBlock * WavePos;       // 256
constexpr int XTile = BlockPos + Wlen;                  // 320
constexpr int Threads = WavesPerBlock * 32;             // 128
constexpr int PStride = 1027;     // odd & ==3 mod 64 -> conflict-free column readers
constexpr int PZeroOff = Kout * PStride;                // 8216
constexpr int PZeroLen = 1024;                          // covers max imm span d*64+w4*4+1
constexpr int SPLen = PZeroOff + PZeroLen;              // 9240 floats (~36.9 KB)
}

// Mid region via implicit-GEMM with V_WMMA_F32_16X16X4_F32.
// Per wave: 4 tiles of D[16 pos x 16 k]; 256 steps (d, w4) each doing
//   B[kk,n] = P[n, d, 4*w4+kk]      (loaded once per step, shared by 4 WMMAs)
//   A_t[m,kk] = x[d, n0+t*16+m+4*w4+kk]
__global__ __launch_bounds__(Threads) void fullscan_mid_wmma(
    const float* __restrict__ xg, const float* __restrict__ Pg,
    float* __restrict__ out)
{
  __shared__ float sP[SPLen];
  __shared__ float sX[Dch * XTile];   // 5120 floats (20 KB)

  const int tid = threadIdx.x;
  const int N0  = blockIdx.x * BlockPos;

  // Stage P: global [k][d][w] -> sP[k*PStride + d*64 + w]
  for (int i = tid; i < Kout * Dch * Wlen; i += Threads) {
    int k  = i >> 10;        // / (16*64)
    int dw = i & 1023;
    sP[k * PStride + dw] = Pg[i];
  }
  // Zero region read by padded N-columns (k = 8..15)
  for (int i = tid; i < PZeroLen; i += Threads) sP[PZeroOff + i] = 0.0f;

  // Stage x tile: sX[d*320 + j] = x[d][min(N0+j, L-1)] (clamp; tail stores masked)
  for (int i = tid; i < Dch * XTile; i += Threads) {
    int d = i / XTile;
    int j = i - d * XTile;
    int gi = N0 + j;
    if (gi > Llen - 1) gi = Llen - 1;
    sX[i] = xg[(size_t)d * Llen + gi];
  }
  __syncthreads();

  const int wave = tid >> 5;   // wave32
  const int lane = tid & 31;
  const int n0 = N0 + wave * WavePos;
  if (n0 >= MidLen) return;    // wave-uniform; EXEC stays all-ones inside

  const int m  = lane & 15;    // A row (position) == C/D column (k)
  const int hi = lane >> 4;    // 0: K-slots {0,1}; 1: K-slots {2,3}
  const int c  = hi * 2;
  const float* xb = &sX[wave * WavePos + m + c];
  const float* pb = (m < Kout) ? &sP[m * PStride + c] : &sP[PZeroOff];

  v8f acc0 = {}, acc1 = {}, acc2 = {}, acc3 = {};
  for (int d = 0; d < Dch; ++d) {
#pragma unroll
    for (int w4 = 0; w4 < 16; ++w4) {
      const int xo = d * XTile + w4 * 4;
      v2f b;
      b.x = pb[d * Wlen + w4 * 4];
      b.y = pb[d * Wlen + w4 * 4 + 1];
      v2f a0, a1, a2, a3;
      a0.x = xb[xo +  0]; a0.y = xb[xo +  1];
      a1.x = xb[xo + 16]; a1.y = xb[xo + 17];
      a2.x = xb[xo + 32]; a2.y = xb[xo + 33];
      a3.x = xb[xo + 48]; a3.y = xb[xo + 49];
      // (neg_a, A, neg_b, B, c_mod, C, reuse_a, reuse_b)
      acc0 = __builtin_amdgcn_wmma_f32_16x16x4_f32(false, a0, false, b, (short)0, acc0, false, false);
      acc1 = __builtin_amdgcn_wmma_f32_16x16x4_f32(false, a1, false, b, (short)0, acc1, false, false);
      acc2 = __builtin_amdgcn_wmma_f32_16x16x4_f32(false, a2, false, b, (short)0, acc2, false, false);
      acc3 = __builtin_amdgcn_wmma_f32_16x16x4_f32(false, a3, false, b, (short)0, acc3, false, false);
    }
  }

  // C/D layout: lane holds N = lane&15; VGPR j holds M = j (lanes 0-15) / j+8 (16-31)
  const int k = m;
  if (k < Kout) {
    const int mrow0 = hi * 8;
    float* outk = out + (size_t)k * Llen + Pad;
#pragma unroll
    for (int j = 0; j < 8; ++j) {
      int n0j = n0 + mrow0 + j;
      if (n0j      < MidLen) outk[n0j]      = acc0[j];
      if (n0j + 16 < MidLen) outk[n0j + 16] = acc1[j];
      if (n0j + 32 < MidLen) outk[n0j + 32] = acc2[j];
      if (n0j + 48 < MidLen) outk[n0j + 48] = acc3[j];
    }
  }
}

// Edge kernel: reproduces the reference's cumsum-of-diagonal edge columns.
// left[k,j]  = sum_{m=0}^{32+j} sum_d x[d,m] * P[k,d,m]          (j = 0..30)
// right[k,j] = sum_{m=j+1}^{63} sum_d x[d, L-64+m] * P[k,d,m]    (j = 0..31)
__global__ void fullscan_edges(const float* __restrict__ xg,
                               const float* __restrict__ Pg,
                               float* __restrict__ out)
{
  int t = blockIdx.x * blockDim.x + threadIdx.x;
  if (t < Kout * Pad) {                       // 248 left outputs
    int k = t / Pad, j = t - k * Pad;
    float s = 0.f;
    for (int mm = 0; mm <= 32 + j; ++mm) {
      float tl = 0.f;
      for (int d = 0; d < Dch; ++d)
        tl += xg[(size_t)d * Llen + mm] * Pg[(k * Dch + d) * Wlen + mm];
      s += tl;
    }
    out[(size_t)k * Llen + j] = s;
  } else if (t < Kout * Pad + Kout * 32) {    // 256 right outputs
    int t2 = t - Kout * Pad;
    int k = t2 / 32, j = t2 - k * 32;
    float s = 0.f;
    for (int mm = j + 1; mm < Wlen; ++mm) {
      for (int d = 0; d < Dch; ++d)
        s += xg[(size_t)d * Llen + (Llen - Wlen + mm)] *
             Pg[(k * Dch + d) * Wlen + mm];
    }
    out[(size_t)k * Llen + (Llen - 32) + j] = s;
  }
}

extern "C" void kernel_launch(void* const* d_in, const int* /*in_sizes*/, int /*n_in*/,
                              void* d_out, int /*out_size*/, void* /*d_ws*/, size_t /*ws_size*/,
                              hipStream_t stream) {
  const float* x = (const float*)d_in[0];   // (16, 1000000) fp32
  const float* P = (const float*)d_in[1];   // (8, 16, 64)  fp32
  float* out = (float*)d_out;               // 8,000,000 fp32

  const int blocks = (MidLen + BlockPos - 1) / BlockPos;  // 3907
  hipLaunchKernelGGL(fullscan_mid_wmma, dim3(blocks), dim3(Threads), 0, stream, x, P, out);
  hipLaunchKernelGGL(fullscan_edges,    dim3(2),      dim3(256),     0, stream, x, P, out);
}